// JointAnfisNet_44873818308905
// MI455X (gfx1250) — compile-verified
//
#include <hip/hip_runtime.h>
#include <hip/hip_bf16.h>
#include <math.h>

// ---------------------------------------------------------------------------
// ANFIS on gfx1250 (wave32). lane = sample; defuzz GEMM via V_WMMA_F32_16X16X4_F32.
// ---------------------------------------------------------------------------

typedef __attribute__((ext_vector_type(2))) float v2f;
typedef __attribute__((ext_vector_type(8))) float v8f;

#if __has_builtin(__builtin_amdgcn_wmma_f32_16x16x4_f32)
#define HAS_WMMA_F32_4 1
#else
#define HAS_WMMA_F32_4 0
#endif

static constexpr int kB      = 16384;
static constexpr int kNIn    = 6;
static constexpr int kNMf    = 5;
static constexpr int kRules  = 2048;
static constexpr int kWG     = 64;     // 2 waves per block, 256 blocks

// ---------------------------------------------------------------------------
// Prep: pack rule indices (6 bytes / rule) + build the WMMA B-operand in its
// exact per-lane VGPR layout (chunk of 4 rules -> B is 4x16, cols>=2 are 0).
// B layout (mirrors C/D striping): VGPR v, lanes 0-15 -> K=v, lanes 16-31 -> K=v+2.
// ---------------------------------------------------------------------------
__global__ void __launch_bounds__(256) anfis_prep(
    const float* __restrict__ oc,        // out_centers [12]
    const int*   __restrict__ irules,    // input_rules [2048*6]
    const int*   __restrict__ orules,    // output_rules [2048*2]
    uint2*       __restrict__ idxpack,   // [2048]
    float*       __restrict__ bmat,      // [512 chunks][2 vgpr][32 lanes]
    float*       __restrict__ owp)       // [2048*2] fallback table
{
  const int t = blockIdx.x * blockDim.x + threadIdx.x;   // exactly 32768 threads
  if (t < kRules) {
    const int* r = irules + t * kNIn;
    unsigned lo = (unsigned)(r[0] & 255) | ((unsigned)(r[1] & 255) << 8) |
                  ((unsigned)(r[2] & 255) << 16) | ((unsigned)(r[3] & 255) << 24);
    unsigned hi = (unsigned)(r[4] & 255) | ((unsigned)(r[5] & 255) << 8);
    idxpack[t] = make_uint2(lo, hi);
    owp[t * 2 + 0] = oc[orules[t * 2 + 0]];
    owp[t * 2 + 1] = oc[orules[t * 2 + 1]];
  }
  // B-operand entry for (chunk, vgpr, lane)
  const int chunk = t >> 6;
  const int pos   = t & 63;
  const int vg    = pos >> 5;
  const int lane  = pos & 31;
  const int half  = lane >> 4;
  const int n     = lane & 15;
  const int k     = chunk * 4 + vg + 2 * half;   // rule index for this K row
  float v = 0.0f;
  if (n < 2) v = oc[orules[k * 2 + n]];
  bmat[t] = v;
}

// ---------------------------------------------------------------------------
// Main kernel
// ---------------------------------------------------------------------------
__global__ void __launch_bounds__(kWG) anfis_main(
    const float* __restrict__ x,        // [B*6]
    const float* __restrict__ mfc,      // [30]
    const float* __restrict__ mfs,      // [30]
    const uint2* __restrict__ idxpack,  // [2048]
    const float* __restrict__ bmat,     // [512*64]
    const float* __restrict__ owp,      // [2048*2] (fallback path only)
    float*       __restrict__ out)      // [B*2]
{
  __shared__ float fuz[kWG * 33];   // stride 33: coprime with 64 banks
  __shared__ float oscr[kWG * 2];

  const int tid    = threadIdx.x;
  const int sample = blockIdx.x * kWG + tid;
  const int lane   = tid & 31;
  const int wave   = tid >> 5;
  const int fb     = tid * 33;

  // ---- Phase 1: fuzzification (30 values per sample into LDS) ----
#pragma unroll
  for (int i = 0; i < kNIn; ++i) {
    const float xi = x[sample * kNIn + i];
#pragma unroll
    for (int m = 0; m < kNMf; ++m) {
      const float z = (xi - mfc[i * kNMf + m]) / mfs[i * kNMf + m];
      fuz[fb + i * kNMf + m] = __expf(-z * z);
    }
  }
  __syncthreads();

  // ---- Phase 2: stream 2048 rules in chunks of 4 (one WMMA K-slab) ----
  float l1 = 0.0f;
#if HAS_WMMA_F32_4
  v8f c0 = {};   // rows = samples (wave*32 + 0..15)
  v8f c1 = {};   // rows = samples (wave*32 + 16..31)
#else
  float acc0 = 0.0f, acc1 = 0.0f;
#endif

  for (int k0 = 0; k0 < kRules; k0 += 4) {
    float w[4];
#pragma unroll
    for (int j = 0; j < 4; ++j) {
      const uint2 u = idxpack[k0 + j];          // uniform -> scalar load, per-lane LDS reads
      const float m0 = fuz[fb + (u.x & 255u)];
      const float m1 = fuz[fb + ((u.x >> 8) & 255u)];
      const float m2 = fuz[fb + ((u.x >> 16) & 255u)];
      const float m3 = fuz[fb + (u.x >> 24)];
      const float m4 = fuz[fb + (u.y & 255u)];
      const float m5 = fuz[fb + ((u.y >> 8) & 255u)];
      const float wv = fminf(fminf(fminf(m0, m1), fminf(m2, m3)), fminf(m4, m5));
      w[j] = wv;
      l1 += fabsf(wv);
#if !HAS_WMMA_F32_4
      acc0 = fmaf(wv, owp[(k0 + j) * 2 + 0], acc0);
      acc1 = fmaf(wv, owp[(k0 + j) * 2 + 1], acc1);
#endif
    }

#if HAS_WMMA_F32_4
    // Redistribute the 4 per-lane weights into the documented 16x4 f32 A layout:
    //   lanes 0-15:  VGPR0=A[M=l][K=0], VGPR1=A[M=l][K=1]
    //   lanes 16-31: VGPR0=A[M=l-16][K=2], VGPR1=A[M=l-16][K=3]
    const int src = lane & 15;
    const float s2 = __shfl(w[2], src, 32);        // w[k2] of lane-16
    const float s3 = __shfl(w[3], src, 32);
    const float s0 = __shfl(w[0], src + 16, 32);   // w[k0] of lane+16
    const float s1 = __shfl(w[1], src + 16, 32);
    const bool lo = lane < 16;
    v2f a0 = { lo ? w[0] : s2, lo ? w[1] : s3 };   // tile of samples 0..15
    v2f a1 = { lo ? s0 : w[2], lo ? s1 : w[3] };   // tile of samples 16..31

    const float* bp = bmat + (k0 >> 2) * 64;       // precomputed B-operand rows
    v2f bv = { bp[lane], bp[32 + lane] };
    __builtin_prefetch(bp + 256, 0, 0);            // global_prefetch_b8

    c0 = __builtin_amdgcn_wmma_f32_16x16x4_f32(false, a0, false, bv, (short)0, c0, false, false);
    c1 = __builtin_amdgcn_wmma_f32_16x16x4_f32(false, a1, false, bv, (short)0, c1, false, false);
#endif
  }

  // ---- Phase 3: pull columns 0/1 of the C tiles back to lane=sample order ----
  float df0, df1;
#if HAS_WMMA_F32_4
  __syncthreads();
  const int col = lane & 15;
  if (col < 2) {
    const int half = lane >> 4;   // C layout: VGPR v -> M=v (lanes 0-15) / M=v+8 (lanes 16-31)
#pragma unroll
    for (int v = 0; v < 8; ++v) {
      oscr[(wave * 32 + v + 8 * half) * 2 + col]      = c0[v];
      oscr[(wave * 32 + 16 + v + 8 * half) * 2 + col] = c1[v];
    }
  }
  __syncthreads();
  df0 = oscr[tid * 2 + 0];
  df1 = oscr[tid * 2 + 1];
#else
  (void)wave; (void)oscr;
  df0 = acc0;
  df1 = acc1;
#endif

  const float inv = 1.0f / fmaxf(l1, 1e-12f);
  out[sample * 2 + 0] = tanhf(df0 * inv) * 4.0f;            // center 0.0
  out[sample * 2 + 1] = tanhf(df1 * inv) * 0.75f + 0.75f;   // center 0.75
}

// ---------------------------------------------------------------------------
// Launch
// ---------------------------------------------------------------------------
extern "C" void kernel_launch(void* const* d_in, const int* in_sizes, int n_in,
                              void* d_out, int out_size, void* d_ws, size_t ws_size,
                              hipStream_t stream) {
  const float* x      = (const float*)d_in[0];
  const float* mfc    = (const float*)d_in[1];
  const float* mfs    = (const float*)d_in[2];
  const float* oc     = (const float*)d_in[3];
  const int*   irules = (const int*)d_in[4];
  const int*   orules = (const int*)d_in[5];
  float* out = (float*)d_out;

  // workspace layout: idxpack (16KB) | bmat (128KB) | owp (16KB)
  uint2* idxpack = (uint2*)d_ws;
  float* bmat    = (float*)((char*)d_ws + kRules * 8);
  float* owp     = (float*)((char*)d_ws + kRules * 8 + (kRules / 4) * 64 * 4);

  anfis_prep<<<128, 256, 0, stream>>>(oc, irules, orules, idxpack, bmat, owp);
  anfis_main<<<kB / kWG, kWG, 0, stream>>>(x, mfc, mfs, idxpack, bmat, owp, out);
}